// ConditionalFeedForward_63376537420019
// MI455X (gfx1250) — compile-verified
//
#include <hip/hip_runtime.h>

#define NUM_EXPERTS 8
#define INTER 2816
#define DIM 1024
#define T_TOK 16
#define TOPK 2
#define NPAIR (T_TOK * TOPK)   // 32 token-expert pairs

typedef __attribute__((ext_vector_type(2))) float v2f;
typedef __attribute__((ext_vector_type(8))) float v8f;
typedef __attribute__((ext_vector_type(4))) int   v4i;

// Workspace layout (bytes):
//   [0     ..   32)  int counts[8]
//   [32    .. 1056)  int lists[8][32]   (pair ids per expert)
//   [4096  ..     )  float H[32][INTER] (gated hidden activations)
#define WS_H_OFFSET 4096

__device__ __forceinline__ float fast_silu_gate(float g, float up) {
    // silu(g) * up with v_exp_f32 + v_rcp_f32 (no IEEE divide chain)
#if defined(__has_builtin) && __has_builtin(__builtin_amdgcn_rcpf)
    float inv = __builtin_amdgcn_rcpf(1.0f + __expf(-g));
#else
    float inv = 1.0f / (1.0f + __expf(-g));
#endif
    return g * inv * up;
}

template <typename V>
__device__ __forceinline__ V nt_load(const V* p) {
    return __builtin_nontemporal_load(p);   // streamed weights: NT hint
}

// ---------------------------------------------------------------------------
// Kernel 0: route the 32 (token, slot) pairs into per-expert lists.
// ---------------------------------------------------------------------------
__global__ void moe_route(const int* __restrict__ eidx, int stride,
                          int* __restrict__ counts, int* __restrict__ lists) {
    if (threadIdx.x == 0 && blockIdx.x == 0) {
        for (int e = 0; e < NUM_EXPERTS; ++e) counts[e] = 0;
        for (int p = 0; p < NPAIR; ++p) {
            int e = eidx[p * stride] & (NUM_EXPERTS - 1);
            lists[e * NPAIR + counts[e]] = p;
            counts[e]++;
        }
    }
}

// ---------------------------------------------------------------------------
// Kernel 1: H[p][n] = silu(x_p . w1[e][n]) * (x_p . w3[e][n])
// GEMM per (expert, m-tile): M<=16 pairs, N=16 per wave (64 per block), K=DIM.
// ---------------------------------------------------------------------------
__global__ __launch_bounds__(128)
void moe_up(const float* __restrict__ x,
            const float* __restrict__ w1,
            const float* __restrict__ w3,
            const int* __restrict__ counts,
            const int* __restrict__ lists,
            float* __restrict__ H) {
    const int e     = blockIdx.y >> 1;
    const int mtile = blockIdx.y & 1;
    const int rows  = counts[e] - mtile * 16;       // valid M rows in this tile
    if (rows <= 0) return;                          // uniform exit, whole block

    __shared__ int   toks[16];
    __shared__ float xs[16 * DIM];                  // gathered token rows, 64 KB

    const int tid = threadIdx.x;
    if (tid < 16) {
        int m = tid;
        int p = (m < rows) ? lists[e * NPAIR + mtile * 16 + m]
                           : lists[e * NPAIR + mtile * 16];   // pad with row 0
        toks[m] = p >> 1;                                     // pair -> token
    }
    __syncthreads();

    // Stage 16 x DIM fp32 activations into LDS, 16B chunks per lane.
#if defined(__has_builtin) && __has_builtin(__builtin_amdgcn_global_load_async_to_lds_b128)
    for (int i = tid; i < 16 * (DIM / 4); i += blockDim.x) {
        int r = i >> 8;          // DIM/4 == 256 float4 per row
        int c = i & 255;
        const float4* src = (const float4*)(x + (size_t)toks[r] * DIM) + c;
        float4*       dst = (float4*)(xs + r * DIM) + c;
        __builtin_amdgcn_global_load_async_to_lds_b128(
            (__attribute__((address_space(1))) v4i*)src,
            (__attribute__((address_space(3))) v4i*)dst,
            /*offset=*/0, /*cpol=*/0);
    }
#if defined(__has_builtin) && __has_builtin(__builtin_amdgcn_s_wait_asynccnt)
    __builtin_amdgcn_s_wait_asynccnt(0);
#else
    asm volatile("s_wait_asynccnt 0x0" ::: "memory");
#endif
#else
    for (int i = tid; i < 16 * (DIM / 4); i += blockDim.x) {
        int r = i >> 8;
        int c = i & 255;
        ((float4*)(xs + r * DIM))[c] =
            ((const float4*)(x + (size_t)toks[r] * DIM))[c];
    }
#endif
    __syncthreads();

    const int wave = tid >> 5;
    const int lane = tid & 31;
    const int half = lane >> 4;       // lanes 16..31 hold K+2,K+3
    const int l16  = lane & 15;
    const int n    = blockIdx.x * 64 + wave * 16 + l16;   // INTER column

    const float* __restrict__ xrow  = xs + l16 * DIM;
    const float* __restrict__ w1row = w1 + ((size_t)e * INTER + n) * DIM;
    const float* __restrict__ w3row = w3 + ((size_t)e * INTER + n) * DIM;

    v8f acc1 = {};
    v8f acc3 = {};
#pragma unroll 4
    for (int k0 = 0; k0 < DIM; k0 += 4) {
        int k   = k0 + 2 * half;
        v2f a  = *(const v2f*)(xrow + k);          // A: 16x4 fp32 frag (LDS)
        v2f b1 = nt_load((const v2f*)(w1row + k)); // B: 4x16 fp32 frag (stream)
        v2f b3 = nt_load((const v2f*)(w3row + k));
        acc1 = __builtin_amdgcn_wmma_f32_16x16x4_f32(
                   false, a, false, b1, (short)0, acc1, false, false);
        acc3 = __builtin_amdgcn_wmma_f32_16x16x4_f32(
                   false, a, false, b3, (short)0, acc3, false, false);
    }

    // D layout: lane L, vgpr r -> M = r + 8*(L>=16), N = L%16.
#pragma unroll
    for (int r = 0; r < 8; ++r) {
        int m = r + half * 8;
        if (m < rows) {
            int p = lists[e * NPAIR + mtile * 16 + m];
            H[(size_t)p * INTER + n] = fast_silu_gate(acc1[r], acc3[r]);
        }
    }
}

// ---------------------------------------------------------------------------
// Kernel 2: out[p][d] = H[p] . w2[e][d]   (K = INTER)
// ---------------------------------------------------------------------------
__global__ __launch_bounds__(128)
void moe_down(const float* __restrict__ H,
              const float* __restrict__ w2,
              const int* __restrict__ counts,
              const int* __restrict__ lists,
              float* __restrict__ out) {
    const int e     = blockIdx.y >> 1;
    const int mtile = blockIdx.y & 1;
    const int rows  = counts[e] - mtile * 16;
    if (rows <= 0) return;

    const int tid  = threadIdx.x;
    const int wave = tid >> 5;
    const int lane = tid & 31;
    const int half = lane >> 4;
    const int l16  = lane & 15;
    const int d    = blockIdx.x * 64 + wave * 16 + l16;   // DIM column

    int pa = (l16 < rows) ? lists[e * NPAIR + mtile * 16 + l16]
                          : lists[e * NPAIR + mtile * 16];
    const float* __restrict__ hrow  = H + (size_t)pa * INTER;
    const float* __restrict__ w2row = w2 + ((size_t)e * DIM + d) * INTER;

    v8f acc = {};
#pragma unroll 4
    for (int k0 = 0; k0 < INTER; k0 += 4) {
        int k  = k0 + 2 * half;
        v2f a = *(const v2f*)(hrow + k);           // H: L2-resident, keep RT
        v2f b = nt_load((const v2f*)(w2row + k));  // w2: streamed once, NT
        acc = __builtin_amdgcn_wmma_f32_16x16x4_f32(
                  false, a, false, b, (short)0, acc, false, false);
    }

#pragma unroll
    for (int r = 0; r < 8; ++r) {
        int m = r + half * 8;
        if (m < rows) {
            int p = lists[e * NPAIR + mtile * 16 + m];
            out[(size_t)p * DIM + d] = acc[r];
        }
    }
}

// ---------------------------------------------------------------------------
extern "C" void kernel_launch(void* const* d_in, const int* in_sizes, int n_in,
                              void* d_out, int out_size, void* d_ws, size_t ws_size,
                              hipStream_t stream) {
    const float* x    = (const float*)d_in[0];
    const int*   eidx = (const int*)d_in[1];
    const float* w1   = (const float*)d_in[2];
    const float* w2   = (const float*)d_in[3];
    const float* w3   = (const float*)d_in[4];

    // expert_indices is int64 in the reference; if the harness hands us the
    // raw 8-byte elements, read the little-endian low words (stride 2).
    int stride = (in_sizes[1] == NPAIR) ? 1 : 2;

    int*   counts = (int*)d_ws;
    int*   lists  = counts + NUM_EXPERTS;
    float* H      = (float*)((char*)d_ws + WS_H_OFFSET);

    moe_route<<<1, 32, 0, stream>>>(eidx, stride, counts, lists);

    dim3 gA(INTER / 64, NUM_EXPERTS * 2);   // 44 x 16
    moe_up<<<gA, 128, 0, stream>>>(x, w1, w3, counts, lists, H);

    dim3 gB(DIM / 64, NUM_EXPERTS * 2);     // 16 x 16
    moe_down<<<gB, 128, 0, stream>>>(H, w2, counts, lists, (float*)d_out);
}